// RoIHeads_41918880809066
// MI455X (gfx1250) — compile-verified
//
#include <hip/hip_runtime.h>
#include <math.h>

// ---------------- problem constants (match reference) ----------------
#define N_PROP        8192
#define NUM_CLASSES   81
#define NUM_FG        (NUM_CLASSES - 1)     // 80
#define NUM_DET       100
#define SCORE_THRESH  0.05f
#define NMS_THRESH    0.5f
#define MIN_SIZE      1.0f
#define BBOX_CLIP     4.135166556742356f    // log(1000/16)

#define NMS_THREADS   256
#define PER_THREAD    (N_PROP / NMS_THREADS) // 32

// ---------------------------------------------------------------------
// Kernel 1: softmax over classes + per-class box decode/clip/threshold.
// Writes class-major tables: scores_ws[fc][n], boxes_ws[fc][n][4]
// so the NMS kernel's async LDS staging is a coalesced linear stream.
// ---------------------------------------------------------------------
__global__ __launch_bounds__(256)
void decode_kernel(const float* __restrict__ proposal,
                   const float* __restrict__ class_logit,
                   const float* __restrict__ box_reg,
                   const int*   __restrict__ image_h,
                   const int*   __restrict__ image_w,
                   float* __restrict__ scores_ws,
                   float* __restrict__ boxes_ws) {
  const int i = blockIdx.x * blockDim.x + threadIdx.x;
  if (i >= N_PROP) return;

  const float W = (float)image_w[0];
  const float H = (float)image_h[0];

  const float* lg = class_logit + (size_t)i * NUM_CLASSES;
  const float* rg = box_reg    + (size_t)i * (NUM_CLASSES * 4);
  // gfx1250: lowers to global_prefetch_b8 (pull the 1.3KB regression row)
  __builtin_prefetch(rg, 0, 1);
  __builtin_prefetch(rg + 128, 0, 1);

  // softmax (2-pass, numerically stable)
  float m = lg[0];
  for (int c = 1; c < NUM_CLASSES; ++c) m = fmaxf(m, lg[c]);
  float ssum = 0.0f;
  for (int c = 0; c < NUM_CLASSES; ++c) ssum += __expf(lg[c] - m);
  const float inv = 1.0f / ssum;

  const float x1p = proposal[i * 4 + 0];
  const float y1p = proposal[i * 4 + 1];
  const float x2p = proposal[i * 4 + 2];
  const float y2p = proposal[i * 4 + 3];
  const float w  = x2p - x1p;
  const float h  = y2p - y1p;
  const float cx = x1p + 0.5f * w;
  const float cy = y1p + 0.5f * h;

  for (int c = 1; c < NUM_CLASSES; ++c) {
    const float dx = rg[c * 4 + 0] * 0.1f;                    // / wx(10)
    const float dy = rg[c * 4 + 1] * 0.1f;                    // / wy(10)
    const float dw = fminf(rg[c * 4 + 2] * 0.2f, BBOX_CLIP);  // / ww(5)
    const float dh = fminf(rg[c * 4 + 3] * 0.2f, BBOX_CLIP);  // / wh(5)

    const float pcx = dx * w + cx;
    const float pcy = dy * h + cy;
    const float pw  = __expf(dw) * w;
    const float ph  = __expf(dh) * h;

    const float x1 = fminf(fmaxf(pcx - 0.5f * pw, 0.0f), W);
    const float y1 = fminf(fmaxf(pcy - 0.5f * ph, 0.0f), H);
    const float x2 = fminf(fmaxf(pcx + 0.5f * pw, 0.0f), W);
    const float y2 = fminf(fmaxf(pcy + 0.5f * ph, 0.0f), H);

    const bool keep = ((x2 - x1) >= MIN_SIZE) && ((y2 - y1) >= MIN_SIZE);
    const float prob = __expf(lg[c] - m) * inv;
    const float s = (prob >= SCORE_THRESH && keep) ? prob : 0.0f;

    const int fc = c - 1;
    scores_ws[(size_t)fc * N_PROP + i] = s;
    float* bo = boxes_ws + ((size_t)fc * N_PROP + i) * 4;
    bo[0] = x1; bo[1] = y1; bo[2] = x2; bo[3] = y2;
  }
}

// ---------------------------------------------------------------------
// Kernel 2: per-class sequential greedy NMS, one workgroup per class.
// Boxes (128KB) + scores (32KB) staged once into LDS via the CDNA5
// async global->LDS path; all 100 iterations then run out of LDS.
// ---------------------------------------------------------------------
#if defined(__gfx1250__) && \
    __has_builtin(__builtin_amdgcn_global_load_async_to_lds_b128) && \
    __has_builtin(__builtin_amdgcn_global_load_async_to_lds_b32)
#define USE_ASYNC_LDS 1
// Parameter types per hipcc diagnostic:
//   b128: (v4i __device__* /*AS1*/, v4i __shared__* /*AS3*/, imm offset, imm cpol)
//   b32 : (int __device__*,         int __shared__*,         imm offset, imm cpol)
typedef int v4i __attribute__((vector_size(16)));
typedef __attribute__((address_space(1))) v4i g_v4i;
typedef __attribute__((address_space(3))) v4i l_v4i;
typedef __attribute__((address_space(1))) int g_i32;
typedef __attribute__((address_space(3))) int l_i32;
#else
#define USE_ASYNC_LDS 0
#endif

__global__ __launch_bounds__(NMS_THREADS)
void nms_kernel(const float* __restrict__ scores_ws,
                const float* __restrict__ boxes_ws,
                float* __restrict__ out) {
  extern __shared__ float smem[];
  float* sBoxes  = smem;                      // N_PROP * 4
  float* sScores = sBoxes + N_PROP * 4;       // N_PROP
  float* redVal  = sScores + N_PROP;          // NMS_THREADS
  float* sBest   = redVal + NMS_THREADS;      // 8 (box4, area, score)
  int*   redIdx  = (int*)(sBest + 8);         // NMS_THREADS

  const int c = blockIdx.x;                   // fg class index 0..79
  const int t = threadIdx.x;
  const float* gScores = scores_ws + (size_t)c * N_PROP;
  const float* gBoxes  = boxes_ws  + (size_t)c * N_PROP * 4;

  // ---- stage class tables into LDS ----
#if USE_ASYNC_LDS
  for (int jj = 0; jj < PER_THREAD; ++jj) {
    const int n = t + jj * NMS_THREADS;
    __builtin_amdgcn_global_load_async_to_lds_b128(
        (g_v4i*)(gBoxes + (size_t)n * 4),
        (l_v4i*)(sBoxes + (size_t)n * 4), 0, 0);
    __builtin_amdgcn_global_load_async_to_lds_b32(
        (g_i32*)(gScores + n),
        (l_i32*)(sScores + n), 0, 0);
  }
  asm volatile("s_wait_asynccnt 0" ::: "memory");
#else
  for (int jj = 0; jj < PER_THREAD; ++jj) {
    const int n = t + jj * NMS_THREADS;
    const float4 b = *(const float4*)(gBoxes + (size_t)n * 4);
    *(float4*)(sBoxes + (size_t)n * 4) = b;
    sScores[n] = gScores[n];
  }
#endif
  __syncthreads();

  // ---- 100 sequential greedy-NMS steps ----
  for (int d = 0; d < NUM_DET; ++d) {
    // per-thread argmax over a strided slice (first-max tie-break order
    // matches jnp.argmax: lowest global index wins on ties)
    float bv = -1.0f;
    int   bi = 0;
    for (int jj = 0; jj < PER_THREAD; ++jj) {
      const int n = t + jj * NMS_THREADS;
      const float v = sScores[n];
      if (v > bv) { bv = v; bi = n; }
    }
    redVal[t] = bv;
    redIdx[t] = bi;
    __syncthreads();

    // block tree reduction (value, then lower index on ties)
    for (int s = NMS_THREADS / 2; s > 0; s >>= 1) {
      if (t < s) {
        const float v2 = redVal[t + s];
        const int   i2 = redIdx[t + s];
        if (v2 > redVal[t] || (v2 == redVal[t] && i2 < redIdx[t])) {
          redVal[t] = v2;
          redIdx[t] = i2;
        }
      }
      __syncthreads();
    }

    if (t == 0) {
      const int   j  = redIdx[0];
      const float sc = redVal[0];
      const float bx1 = sBoxes[j * 4 + 0];
      const float by1 = sBoxes[j * 4 + 1];
      const float bx2 = sBoxes[j * 4 + 2];
      const float by2 = sBoxes[j * 4 + 3];
      sBest[0] = bx1; sBest[1] = by1; sBest[2] = bx2; sBest[3] = by2;
      sBest[4] = (bx2 - bx1) * (by2 - by1);
      sBest[5] = sc;
      // outputs: boxes | scores | labels | valid (concatenated, float)
      const int o = c * NUM_DET + d;
      out[o * 4 + 0] = bx1;
      out[o * 4 + 1] = by1;
      out[o * 4 + 2] = bx2;
      out[o * 4 + 3] = by2;
      out[NUM_FG * NUM_DET * 4 + o]                         = sc;
      out[NUM_FG * NUM_DET * 4 + NUM_FG * NUM_DET + o]      = (float)(c + 1);
      out[NUM_FG * NUM_DET * 4 + 2 * NUM_FG * NUM_DET + o]  = (sc > 0.0f) ? 1.0f : 0.0f;
    }
    __syncthreads();

    const float bx1 = sBest[0], by1 = sBest[1];
    const float bx2 = sBest[2], by2 = sBest[3];
    const float ba  = sBest[4], bsc = sBest[5];
    if (bsc > 0.0f) {   // reference: suppress only when pick is valid
      for (int jj = 0; jj < PER_THREAD; ++jj) {
        const int n = t + jj * NMS_THREADS;
        const float x1 = sBoxes[n * 4 + 0];
        const float y1 = sBoxes[n * 4 + 1];
        const float x2 = sBoxes[n * 4 + 2];
        const float y2 = sBoxes[n * 4 + 3];
        float iw = fminf(bx2, x2) - fmaxf(bx1, x1);
        float ih = fminf(by2, y2) - fmaxf(by1, y1);
        iw = fmaxf(iw, 0.0f);
        ih = fmaxf(ih, 0.0f);
        const float inter = iw * ih;
        const float a2 = (x2 - x1) * (y2 - y1);
        const float iou = inter / (ba + a2 - inter + 1e-12f);
        if (iou > NMS_THRESH) sScores[n] = 0.0f;  // also zeroes pick (iou==1)
      }
    }
    __syncthreads();
  }
}

// ---------------------------------------------------------------------
extern "C" void kernel_launch(void* const* d_in, const int* in_sizes, int n_in,
                              void* d_out, int out_size, void* d_ws, size_t ws_size,
                              hipStream_t stream) {
  (void)in_sizes; (void)n_in; (void)out_size; (void)ws_size;
  const float* proposal    = (const float*)d_in[0];
  const float* class_logit = (const float*)d_in[1];
  const float* box_reg     = (const float*)d_in[2];
  const int*   image_h     = (const int*)d_in[3];
  const int*   image_w     = (const int*)d_in[4];

  float* scores_ws = (float*)d_ws;                               // 80*8192 f32
  float* boxes_ws  = scores_ws + (size_t)NUM_FG * N_PROP;        // 80*8192*4 f32
  float* out       = (float*)d_out;

  decode_kernel<<<N_PROP / 256, 256, 0, stream>>>(
      proposal, class_logit, box_reg, image_h, image_w, scores_ws, boxes_ws);

  const size_t lds_bytes =
      (size_t)(N_PROP * 4 + N_PROP + NMS_THREADS + 8 + NMS_THREADS) * sizeof(float);
  nms_kernel<<<NUM_FG, NMS_THREADS, lds_bytes, stream>>>(scores_ws, boxes_ws, out);
}